// mLSTMCell_10531259810151
// MI455X (gfx1250) — compile-verified
//
#include <hip/hip_runtime.h>
#include <cstddef>

typedef float v2f __attribute__((ext_vector_type(2)));
typedef float v4f __attribute__((ext_vector_type(4)));
typedef float v8f __attribute__((ext_vector_type(8)));

#define BB 64
#define II 1024
#define HH 1024
#define KGATE (II + HH)

// ---------------------------------------------------------------------------
// Build combined = [x, h_prev]  : [64, 2048]
// ---------------------------------------------------------------------------
__global__ void concat_kernel(const float* __restrict__ x,
                              const float* __restrict__ h,
                              float* __restrict__ comb) {
  int t = blockIdx.x * blockDim.x + threadIdx.x;
  int total = BB * KGATE;
  if (t >= total) return;
  int b = t / KGATE;
  int c = t - b * KGATE;
  comb[t] = (c < II) ? x[b * II + c] : h[b * HH + (c - II)];
}

// ---------------------------------------------------------------------------
// WMMA fp32 GEMM:  out[M,N] = act(A[M,K] @ W[K,N] + bias[N])
// One wave (32 lanes) per 16x16 output tile, V_WMMA_F32_16X16X4_F32 over K.
// act: 0 = identity, 1 = sigmoid, 2 = exp
// A layout (16x4 f32): lanes 0-15 -> M=lane, K={0,1}; lanes 16-31 -> K={2,3}
// B layout (4x16 f32): VGPR v, lane l -> K = v + 2*(l>>4), N = l&15
// C/D layout: VGPR r, lane l -> M = r + 8*(l>>4), N = l&15
// ---------------------------------------------------------------------------
__global__ void wmma_gemm_bias_act(const float* __restrict__ A,
                                   const float* __restrict__ W,
                                   const float* __restrict__ bias,
                                   float* __restrict__ out,
                                   int M, int N, int K, int act) {
  const int lane  = threadIdx.x & 31;
  const int m0    = blockIdx.y * 16;
  const int n0    = blockIdx.x * 16;
  const int lo16  = lane & 15;
  const int half  = lane >> 4;      // 0 or 1
  const int kgrp  = half * 2;       // 0 or 2
  const int mrow  = m0 + lo16;
  const int ncol  = n0 + lo16;

  v8f acc = {};
  const float* Arow = A + (size_t)mrow * K;
  for (int k0 = 0; k0 < K; k0 += 4) {
    v2f a = *(const v2f*)(Arow + k0 + kgrp);
    v2f b;
    b.x = W[(size_t)(k0 + kgrp + 0) * N + ncol];
    b.y = W[(size_t)(k0 + kgrp + 1) * N + ncol];
    acc = __builtin_amdgcn_wmma_f32_16x16x4_f32(
        /*neg_a=*/false, a, /*neg_b=*/false, b,
        /*c_mod=*/(short)0, acc, /*reuse_a=*/false, /*reuse_b=*/false);
  }

  const float bv = bias[ncol];
#pragma unroll
  for (int r = 0; r < 8; ++r) {
    int row = m0 + r + 8 * half;
    float v = acc[r] + bv;
    if (act == 1)      v = 1.0f / (1.0f + __expf(-v));
    else if (act == 2) v = __expf(v);
    out[(size_t)row * N + (n0 + lo16)] = v;
  }
}

// ---------------------------------------------------------------------------
// n_t = f * n_prev + i * e        (e = exp(k@We + be), precomputed)
// ---------------------------------------------------------------------------
__global__ void n_kernel(const float* __restrict__ fg,
                         const float* __restrict__ ig,
                         const float* __restrict__ e,
                         const float* __restrict__ n_prev,
                         float* __restrict__ n_out) {
  int t = blockIdx.x * blockDim.x + threadIdx.x;
  if (t < BB * HH) n_out[t] = fg[t] * n_prev[t] + ig[t] * e[t];
}

// ---------------------------------------------------------------------------
// Fused, bandwidth-bound kernel (the 536 MB of HBM traffic lives here):
//   C_t[b,i,j] = f[b,i]*C_prev[b,i,j] + i[b,i]*v[b,i]*k[b,j]
//   h_t[b,i]   = o[b,i] * (sum_j C_t[b,i,j]*q[b,j]) / n_t[b,i]
// One wave32 per (b,i) row; NT b128 streams for C_prev/C_t, cached k/q.
// ---------------------------------------------------------------------------
__global__ __launch_bounds__(256)
void c_update_kernel(const float* __restrict__ Cprev,
                     const float* __restrict__ fg,
                     const float* __restrict__ ig,
                     const float* __restrict__ og,
                     const float* __restrict__ kk,
                     const float* __restrict__ vv,
                     const float* __restrict__ qq,
                     const float* __restrict__ nt,
                     float* __restrict__ Cout,
                     float* __restrict__ hout) {
  const int lane = threadIdx.x & 31;
  const int wave = blockIdx.x * (blockDim.x >> 5) + (threadIdx.x >> 5);
  const int b = wave >> 10;          // wave / 1024
  const size_t rowoff = (size_t)wave * HH;

  const float f = fg[wave];
  const float a = ig[wave] * vv[wave];

  const v4f* __restrict__ cp = (const v4f*)(Cprev + rowoff);
  v4f* __restrict__ co       = (v4f*)(Cout + rowoff);
  const v4f* __restrict__ kp = (const v4f*)(kk + (size_t)b * HH);
  const v4f* __restrict__ qp = (const v4f*)(qq + (size_t)b * HH);

  float acc = 0.0f;
#pragma unroll
  for (int it = 0; it < (HH / 4) / 32; ++it) {   // 8 iterations
    const int idx = it * 32 + lane;
    v4f c4 = __builtin_nontemporal_load(&cp[idx]);
    v4f k4 = kp[idx];
    v4f q4 = qp[idx];
    v4f ct;
    ct.x = __builtin_fmaf(f, c4.x, a * k4.x);
    ct.y = __builtin_fmaf(f, c4.y, a * k4.y);
    ct.z = __builtin_fmaf(f, c4.z, a * k4.z);
    ct.w = __builtin_fmaf(f, c4.w, a * k4.w);
    __builtin_nontemporal_store(ct, &co[idx]);
    acc = __builtin_fmaf(ct.x, q4.x, acc);
    acc = __builtin_fmaf(ct.y, q4.y, acc);
    acc = __builtin_fmaf(ct.z, q4.z, acc);
    acc = __builtin_fmaf(ct.w, q4.w, acc);
  }
#pragma unroll
  for (int off = 16; off > 0; off >>= 1)
    acc += __shfl_down(acc, off, 32);
  if (lane == 0)
    hout[wave] = og[wave] * acc / nt[wave];
}

// ---------------------------------------------------------------------------
extern "C" void kernel_launch(void* const* d_in, const int* in_sizes, int n_in,
                              void* d_out, int out_size, void* d_ws, size_t ws_size,
                              hipStream_t stream) {
  const float* x      = (const float*)d_in[0];
  const float* h_prev = (const float*)d_in[1];
  const float* C_prev = (const float*)d_in[2];
  const float* n_prev = (const float*)d_in[3];
  const float* Wi = (const float*)d_in[4];  const float* bi = (const float*)d_in[5];
  const float* Wf = (const float*)d_in[6];  const float* bf = (const float*)d_in[7];
  const float* Wo = (const float*)d_in[8];  const float* bo = (const float*)d_in[9];
  const float* Wk = (const float*)d_in[10]; const float* bk = (const float*)d_in[11];
  const float* Wv = (const float*)d_in[12]; const float* bv = (const float*)d_in[13];
  const float* Wq = (const float*)d_in[14]; const float* bq = (const float*)d_in[15];
  const float* We = (const float*)d_in[16]; const float* be = (const float*)d_in[17];

  float* out  = (float*)d_out;
  float* h_t  = out;                                   // [64,1024]
  float* C_t  = out + (size_t)BB * HH;                 // [64,1024,1024]
  float* n_t  = out + (size_t)BB * HH + (size_t)BB * HH * HH;

  float* ws   = (float*)d_ws;
  float* comb = ws;                         // 64*2048
  float* ig   = comb + BB * KGATE;          // 64*1024 each below
  float* fg   = ig   + BB * HH;
  float* og   = fg   + BB * HH;
  float* kbuf = og   + BB * HH;
  float* vbuf = kbuf + BB * HH;
  float* qbuf = vbuf + BB * HH;
  float* ebuf = qbuf + BB * HH;

  // 1) combined = [x, h_prev]
  concat_kernel<<<(BB * KGATE + 255) / 256, 256, 0, stream>>>(x, h_prev, comb);

  // 2) WMMA GEMMs (one wave per 16x16 tile)
  dim3 blk(32);
  dim3 grdGate(HH / 16, BB / 16);   // (64, 4)
  wmma_gemm_bias_act<<<grdGate, blk, 0, stream>>>(comb, Wi, bi, ig, BB, HH, KGATE, 1);
  wmma_gemm_bias_act<<<grdGate, blk, 0, stream>>>(comb, Wf, bf, fg, BB, HH, KGATE, 1);
  wmma_gemm_bias_act<<<grdGate, blk, 0, stream>>>(comb, Wo, bo, og, BB, HH, KGATE, 1);
  wmma_gemm_bias_act<<<grdGate, blk, 0, stream>>>(x,    Wk, bk, kbuf, BB, HH, II, 0);
  wmma_gemm_bias_act<<<grdGate, blk, 0, stream>>>(x,    Wv, bv, vbuf, BB, HH, II, 0);
  wmma_gemm_bias_act<<<grdGate, blk, 0, stream>>>(h_prev, Wq, bq, qbuf, BB, HH, HH, 0);
  // e = exp(k @ We + be)   (depends on kbuf)
  wmma_gemm_bias_act<<<grdGate, blk, 0, stream>>>(kbuf, We, be, ebuf, BB, HH, HH, 2);

  // 3) n_t = f*n_prev + i*e
  n_kernel<<<(BB * HH + 255) / 256, 256, 0, stream>>>(fg, ig, ebuf, n_prev, n_t);

  // 4) fused C_t update + h_t matvec: one wave per (b,i) row, 65536 waves
  const int waves = BB * HH;
  c_update_kernel<<<waves / 8, 256, 0, stream>>>(C_prev, fg, ig, og, kbuf, vbuf,
                                                 qbuf, n_t, C_t, h_t);
}